// BoundaryLoss_15805479649582
// MI455X (gfx1250) — compile-verified
//
#include <hip/hip_runtime.h>
#include <hip/hip_bf16.h>
#include <math.h>

#define B_   8
#define C_   21
#define H_   512
#define W_   512
#define TW   32
#define TH   8
#define HALO 2
#define SW   (TW + 2*HALO)   // 36
#define SH   (TH + 2*HALO)   // 12

#if defined(__AMDGCN__) && __has_builtin(__builtin_amdgcn_global_load_async_to_lds_b32) && __has_builtin(__builtin_amdgcn_s_wait_asynccnt)
#define USE_ASYNC 1
#else
#define USE_ASYNC 0
#endif

#if USE_ASYNC
typedef __attribute__((address_space(1))) int g_int;
typedef __attribute__((address_space(3))) int l_int;
__device__ __forceinline__ void async_copy_f32(const float* g, float* l) {
  __builtin_amdgcn_global_load_async_to_lds_b32(
      (g_int*)(float*)g,
      (l_int*)l,
      /*offset=*/0, /*cpol=*/0);
}
#endif

__global__ __launch_bounds__(256) void BL_main(
    const float* __restrict__ pred, const int* __restrict__ target,
    float* __restrict__ accum)
{
  __shared__ int   t_s[SH * SW];
  __shared__ float p_s[2 * 256];

  const int b  = blockIdx.z;
  const int h0 = blockIdx.y * TH;
  const int w0 = blockIdx.x * TW;
  const int tx = threadIdx.x;            // 0..31 == lane id (wave32)
  const int ty = threadIdx.y;            // 0..7  == wave id in block
  const int tid = ty * TW + tx;

  // ---- stage target halo tile into LDS ----
  const int* tgt_b = target + (size_t)b * H_ * W_;
  for (int i = tid; i < SH * SW; i += 256) {
    int sy = i / SW, sx = i % SW;
    int gh = h0 - HALO + sy;
    int gw = w0 - HALO + sx;
    int v = -1;
    if ((unsigned)gh < (unsigned)H_ && (unsigned)gw < (unsigned)W_)
      v = tgt_b[gh * W_ + gw];
    t_s[i] = v;
  }
  __syncthreads();

  const int h = h0 + ty;
  const int w = w0 + tx;
  const int t = t_s[(ty + HALO) * SW + (tx + HALO)];

  // ---- boundary mask: >=2 distinct in-bounds values in 5x5 window ----
  int vmin = t, vmax = t;
  #pragma unroll
  for (int dy = -HALO; dy <= HALO; ++dy) {
    int gh = h + dy;
    if ((unsigned)gh >= (unsigned)H_) continue;
    #pragma unroll
    for (int dx = -HALO; dx <= HALO; ++dx) {
      int gw = w + dx;
      if ((unsigned)gw >= (unsigned)W_) continue;
      int v = t_s[(ty + HALO + dy) * SW + (tx + HALO + dx)];
      vmin = min(vmin, v);
      vmax = max(vmax, v);
    }
  }
  const float mask = (vmin != vmax) ? 1.0f : 0.0f;

  // ---- online softmax CE over 21 classes ----
  const size_t HWsz = (size_t)H_ * W_;
  const float* pp = pred + ((size_t)b * C_ * H_ + h) * W_ + w;

  float m = -INFINITY, s = 0.0f, pt = 0.0f;

#if USE_ASYNC
  // Per-wave async double-buffer: each lane DMAs its own class value into
  // its private LDS slot; the wave waits only on its own ASYNCcnt and reads
  // only LDS it wrote -> no workgroup barriers in the pipeline.
  volatile float* ps = (volatile float*)p_s;
  const float* src = pp;
  async_copy_f32(src, &p_s[tid]);                        // class 0 -> buf 0
  #pragma unroll 4
  for (int c = 0; c < C_ - 1; ++c) {
    src += HWsz;
    async_copy_f32(src, &p_s[((c + 1) & 1) * 256 + tid]); // prefetch c+1
    __builtin_amdgcn_s_wait_asynccnt(1);                  // buf c ready
    __asm__ __volatile__("" ::: "memory");
    float p = ps[(c & 1) * 256 + tid];
    if (c == t) pt = p;
    float nm = fmaxf(m, p);
    s = s * __expf(m - nm) + __expf(p - nm);
    m = nm;
  }
  __builtin_amdgcn_s_wait_asynccnt(0);                    // last buffer ready
  __asm__ __volatile__("" ::: "memory");
  {
    const int c = C_ - 1;
    float p = ps[(c & 1) * 256 + tid];
    if (c == t) pt = p;
    float nm = fmaxf(m, p);
    s = s * __expf(m - nm) + __expf(p - nm);
    m = nm;
  }
#else
  #pragma unroll 4
  for (int c = 0; c < C_; ++c) {
    float p = pp[(size_t)c * HWsz];
    if (c == t) pt = p;
    float nm = fmaxf(m, p);
    s = s * __expf(m - nm) + __expf(p - nm);
    m = nm;
  }
#endif

  const float ce  = (m + __logf(s)) - pt;
  const float wce = ce * mask;

  // ---- wave32 reduction, then 2 atomics per wave ----
  float msum = mask, wsum = wce;
  #pragma unroll
  for (int off = 16; off > 0; off >>= 1) {
    msum += __shfl_xor(msum, off, 32);
    wsum += __shfl_xor(wsum, off, 32);
  }
  if (tx == 0) {
    atomicAdd(&accum[2 * b + 0], msum);
    atomicAdd(&accum[2 * b + 1], wsum);
  }
}

__global__ void BL_zero_ws(float* a) {
  if (threadIdx.x < 2 * B_) a[threadIdx.x] = 0.0f;
}

__global__ void BL_finalize(const float* __restrict__ accum,
                            float* __restrict__ out)
{
  if (threadIdx.x == 0 && blockIdx.x == 0) {
    float total = 0.0f;
    #pragma unroll
    for (int b = 0; b < B_; ++b) {
      float msum = accum[2 * b + 0];
      float wsum = accum[2 * b + 1];
      float per  = (msum > 0.0f) ? (wsum / fmaxf(msum, 1.0f))
                                 : (wsum / (float)(H_ * W_));
      total += per;
    }
    out[0] = total / (float)B_;
  }
}

extern "C" void kernel_launch(void* const* d_in, const int* in_sizes, int n_in,
                              void* d_out, int out_size, void* d_ws, size_t ws_size,
                              hipStream_t stream) {
  const float* pred   = (const float*)d_in[0];
  const int*   target = (const int*)d_in[1];
  float*       accum  = (float*)d_ws;
  float*       out    = (float*)d_out;

  BL_zero_ws<<<1, 32, 0, stream>>>(accum);

  dim3 grid(W_ / TW, H_ / TH, B_);   // (16, 64, 8)
  dim3 block(TW, TH, 1);             // 256 threads = 8 waves (wave32)
  BL_main<<<grid, block, 0, stream>>>(pred, target, accum);

  BL_finalize<<<1, 32, 0, stream>>>(accum, out);
}